// MaxPool4d_13168369729619
// MI455X (gfx1250) — compile-verified
//
#include <hip/hip_runtime.h>

// 4D max pooling over trailing (K,D,H,W) dims of x[N,C,K,D,H,W],
// window=2, stride=2, VALID. Pure streaming (no reuse) -> direct
// wide NT loads are bandwidth-optimal on MI455X (23.3 TB/s floor ~12us).
//
// Block decomposition: one 256-thread block covers the full (ho=32 x wq=8)
// plane for a fixed (n,c,ko,dq), so all higher coordinates come from
// blockIdx.x and are wave-uniform -> SALU address math + GVS addressing
// (scalar 64-bit base + small per-lane VGPR offset + immediate IOFFSET).

typedef float v4f __attribute__((ext_vector_type(4)));

#define N_  4
#define C_  16
#define K_  16
#define D_  16
#define H_  64
#define W_  64
#define KO  (K_ / 2)
#define DO_ (D_ / 2)
#define HO  (H_ / 2)
#define WO  (W_ / 2)

__global__ __launch_bounds__(256)
void maxpool4d_kernel(const float* __restrict__ x,
                      float* __restrict__ out) {
    // ---- per-lane coordinates (VALU, cheap) ----
    const int t  = threadIdx.x;        // 0..255
    const int wq = t & 7;              // 8 groups of 4 outputs along W (WO=32)
    const int ho = t >> 3;             // 0..31

    // ---- block-uniform coordinates (SALU) ----
    int b = blockIdx.x;                // 0..4095
    const int dq = b & 7;  b >>= 3;    // DO = 8
    const int ko = b & 7;  b >>= 3;    // KO = 8
    const int c  = b & 15; b >>= 4;    // C  = 16
    const int n  = b;                  // N  = 4

    const int  sH = W_;                    // 64
    const int  sD = H_ * W_;               // 4096
    const int  sK = D_ * H_ * W_;          // 65536
    const long sC = (long)K_ * sK;         // 1,048,576
    const long sN = (long)C_ * sC;         // 16,777,216

    // Uniform base (SGPRs) + per-lane offset (single VGPR).
    const float* base = x + (long)n * sN + (long)c * sC
                          + (long)(2 * ko) * sK
                          + (long)(2 * dq) * sD;
    const int lane_off = 2 * ho * sH + 8 * wq;   // < 2^14 elements

    const float* p = base + lane_off;

    float m0, m1, m2, m3;
    #pragma unroll
    for (int r = 0; r < 8; ++r) {
        const int kk = r >> 2, dd = (r >> 1) & 1, hh = r & 1;
        // All row offsets are immediates: kk*262144 + dd*16384 + hh*256 (+16) bytes < 2^23.
        const float* row = p + kk * sK + dd * sD + hh * sH;
        // Non-temporal: 256 MiB stream, consumed once -> don't thrash the 192MB L2.
        v4f a = __builtin_nontemporal_load((const v4f*)row);
        v4f b4 = __builtin_nontemporal_load((const v4f*)(row + 4));
        if (r == 0) {
            m0 = fmaxf(a.x, a.y);
            m1 = fmaxf(a.z, a.w);
            m2 = fmaxf(b4.x, b4.y);
            m3 = fmaxf(b4.z, b4.w);
        } else {
            // fmax(m, fmax(p,q)) -> single v_max3_num_f32 on gfx1250
            m0 = fmaxf(m0, fmaxf(a.x, a.y));
            m1 = fmaxf(m1, fmaxf(a.z, a.w));
            m2 = fmaxf(m2, fmaxf(b4.x, b4.y));
            m3 = fmaxf(m3, fmaxf(b4.z, b4.w));
        }
    }

    // Output: uniform part from block coords; per-lane part is exactly 4*t
    // elements (ho*WO + 4*wq == 4*t), i.e. 16*t bytes -> trivial addressing.
    float* o = out
             + ((long)(((n * C_ + c) * KO + ko) * DO_ + dq)) * (HO * WO)
             + 4 * t;
    v4f v = {m0, m1, m2, m3};
    __builtin_nontemporal_store(v, (v4f*)o);
}

extern "C" void kernel_launch(void* const* d_in, const int* in_sizes, int n_in,
                              void* d_out, int out_size, void* d_ws, size_t ws_size,
                              hipStream_t stream) {
    const float* x = (const float*)d_in[0];
    float* out = (float*)d_out;

    // Grid exactly tiles the output: 4096 blocks x 256 threads x 4 outputs
    // = 4,194,304 = out_size. No tail, no bounds check.
    (void)in_sizes; (void)n_in; (void)out_size; (void)d_ws; (void)ws_size;
    dim3 block(256);                 // 8 wave32s per workgroup
    dim3 grid(N_ * C_ * KO * DO_);   // 4096

    maxpool4d_kernel<<<grid, block, 0, stream>>>(x, out);
}